// feature_extraction_3040836845602
// MI455X (gfx1250) — compile-verified
//
#include <hip/hip_runtime.h>

typedef __attribute__((ext_vector_type(2))) float v2f;
typedef __attribute__((ext_vector_type(8))) float v8f;

#define BB 4
#define NN 4096
#define KK 17   // K+1 neighbors including self

#if defined(__has_builtin)
#if __has_builtin(__builtin_amdgcn_global_load_async_to_lds_b32) && \
    __has_builtin(__builtin_amdgcn_s_wait_asynccnt)
#define USE_ASYNC_LDS 1
#endif
#endif

typedef __attribute__((address_space(1))) int as1_int;
typedef __attribute__((address_space(3))) int as3_int;

__device__ __forceinline__ v8f splat8(float x) {
  v8f r;
#pragma unroll
  for (int i = 0; i < 8; ++i) r[i] = x;
  return r;
}

__device__ __forceinline__ v8f wmma4(v2f a, v2f b, v8f c) {
  return __builtin_amdgcn_wmma_f32_16x16x4_f32(false, a, false, b,
                                               (short)0, c, false, false);
}

// ---------------------------------------------------------------------------
// c1: l0[b][n][o] = relu(b[o] + sum_c x[b][c][n] * w[o][c]),  3 -> 24
// ---------------------------------------------------------------------------
__global__ __launch_bounds__(256) void k_c1(const float* __restrict__ x,
                                            const float* __restrict__ w,
                                            const float* __restrict__ bias,
                                            float* __restrict__ l0) {
  int t = blockIdx.x * 256 + threadIdx.x;
  if (t >= BB * NN) return;
  int b = t / NN, n = t % NN;
  const float* xb = x + (size_t)b * 3 * NN;
  float p0 = xb[0 * NN + n], p1 = xb[1 * NN + n], p2 = xb[2 * NN + n];
  float* o = l0 + (size_t)t * 24;
#pragma unroll
  for (int c = 0; c < 24; ++c) {
    float a = bias[c] + w[c * 3 + 0] * p0 + w[c * 3 + 1] * p1 + w[c * 3 + 2] * p2;
    o[c] = fmaxf(a, 0.f);
  }
}

// ---------------------------------------------------------------------------
// squared norms of feature rows
// ---------------------------------------------------------------------------
__global__ __launch_bounds__(256) void k_sqnorm(const float* __restrict__ f,
                                                float* __restrict__ sq, int C) {
  int i = blockIdx.x * 256 + threadIdx.x;
  if (i >= NN) return;
  const float* r = f + (size_t)i * C;
  float a = 0.f;
  for (int c = 0; c < C; ++c) a += r[c] * r[c];
  sq[i] = a;
}

// ---------------------------------------------------------------------------
// Gram matrix G[i][j] = <f_i,f_j> via V_WMMA_F32_16X16X4_F32.
// 8 waves/block; each wave computes a 16x64 strip (4 j-tiles, A reused 4x).
// grid = (8, 256): j-strips, i-tiles.
// ---------------------------------------------------------------------------
__global__ __launch_bounds__(256) void k_gram(const float* __restrict__ f,
                                              float* __restrict__ G, int C) {
  int wave = threadIdx.x >> 5;
  int lane = threadIdx.x & 31;
  int ibase = blockIdx.y * 16;
  int jt0 = (blockIdx.x * 8 + wave) * 4;  // first of 4 j-tiles
  int row  = lane & 15;
  int koff = (lane >> 4) << 1;

  const float* arow = f + (size_t)(ibase + row) * C;

  v8f acc[4];
#pragma unroll
  for (int t = 0; t < 4; ++t) acc[t] = splat8(0.f);

  for (int k0 = 0; k0 < C; k0 += 4) {
    v2f a;
    a.x = arow[k0 + koff];
    a.y = arow[k0 + koff + 1];
#pragma unroll
    for (int t = 0; t < 4; ++t) {
      const float* brow = f + (size_t)((jt0 + t) * 16 + row) * C;
      v2f b;
      b.x = brow[k0 + koff];
      b.y = brow[k0 + koff + 1];
      acc[t] = wmma4(a, b, acc[t]);
    }
  }
  int col   = lane & 15;
  int rbase = ibase + ((lane >> 4) << 3);
#pragma unroll
  for (int t = 0; t < 4; ++t)
#pragma unroll
    for (int v = 0; v < 8; ++v)
      G[(size_t)(rbase + v) * NN + ((jt0 + t) * 16 + col)] = acc[t][v];
}

// ---------------------------------------------------------------------------
// top-17 smallest of d_j = sq[j] - 2*G[j][i]  (G symmetric -> coalesced reads)
// ---------------------------------------------------------------------------
__global__ __launch_bounds__(256) void k_topk(const float* __restrict__ G,
                                              const float* __restrict__ sq,
                                              int* __restrict__ idx) {
  int i = blockIdx.x * 256 + threadIdx.x;
  if (i >= NN) return;
  float v[KK];
  int ix[KK];
#pragma unroll
  for (int s = 0; s < KK; ++s) { v[s] = 3.4e38f; ix[s] = 0; }
  for (int j = 0; j < NN; ++j) {
    if (j + 64 < NN) __builtin_prefetch(G + (size_t)(j + 64) * NN + i, 0, 1);
    float d = sq[j] - 2.0f * G[(size_t)j * NN + i];
    if (d < v[KK - 1]) {
#pragma unroll
      for (int s = KK - 1; s > 0; --s) {
        if (d < v[s]) {
          if (d < v[s - 1]) { v[s] = v[s - 1]; ix[s] = ix[s - 1]; }
          else              { v[s] = d;        ix[s] = j; }
        }
      }
      if (d < v[0]) { v[0] = d; ix[0] = j; }
    }
  }
#pragma unroll
  for (int s = 0; s < KK; ++s) idx[i * KK + s] = ix[s];
}

// ---------------------------------------------------------------------------
// WMMA pointwise linear + relu, Cout = 48 (3 j-tiles), rows = NN.
// grid = (3, 32), block = 256 (8 waves -> 256 i-tiles).
// ---------------------------------------------------------------------------
__global__ __launch_bounds__(256) void k_lin48(const float* __restrict__ src,
                                               const float* __restrict__ w,
                                               const float* __restrict__ bias,
                                               float* __restrict__ dst, int Cin) {
  int wave = threadIdx.x >> 5;
  int lane = threadIdx.x & 31;
  int it = blockIdx.y * 8 + wave;
  int jbase = blockIdx.x * 16;
  int row  = lane & 15;
  int koff = (lane >> 4) << 1;

  const float* arow = src + (size_t)(it * 16 + row) * Cin;
  const float* wrow = w + (size_t)(jbase + row) * Cin;  // B[k][n] = W[n][k]

  v8f acc = splat8(0.f);
  for (int k0 = 0; k0 < Cin; k0 += 4) {
    v2f a, b;
    a.x = arow[k0 + koff];
    a.y = arow[k0 + koff + 1];
    b.x = wrow[k0 + koff];
    b.y = wrow[k0 + koff + 1];
    acc = wmma4(a, b, acc);
  }
  int col   = lane & 15;
  int rbase = it * 16 + ((lane >> 4) << 3);
  float bo = bias[jbase + col];
#pragma unroll
  for (int v = 0; v < 8; ++v)
    dst[(size_t)(rbase + v) * 48 + jbase + col] = fmaxf(acc[v] + bo, 0.f);
}

// ---------------------------------------------------------------------------
// one edge-MLP layer as WMMA: h(16x24) = A(16xK) x W^T(Kx24), K = Kdim.
// A read from per-wave LDS tile Y (row stride LR), B from LDS weights at wOff.
// Output cols padded to 32 (two 16-wide D tiles); tile1 cols >= 24 are zero-B.
// Optionally relu + store h back into Y at storeOff, and track running max.
// ---------------------------------------------------------------------------
__device__ __forceinline__ void edge_layer(float* Y, const float* ws, int wOff,
                                           int bOff, int cbase, int Kdim, int LR,
                                           int row16, int half, int koff,
                                           bool relu, bool store, int storeOff,
                                           v8f& mxa, v8f& mxb) {
  v8f c0 = splat8(0.f), c1 = splat8(0.f);
  for (int k0 = 0; k0 < Kdim; k0 += 4) {
    v2f a, b0, b1;
    a.x = Y[row16 * LR + cbase + k0 + koff];
    a.y = Y[row16 * LR + cbase + k0 + koff + 1];
    b0.x = ws[wOff + row16 * Kdim + k0 + koff];
    b0.y = ws[wOff + row16 * Kdim + k0 + koff + 1];
    if (row16 < 8) {
      b1.x = ws[wOff + (16 + row16) * Kdim + k0 + koff];
      b1.y = ws[wOff + (16 + row16) * Kdim + k0 + koff + 1];
    } else {
      b1.x = 0.f; b1.y = 0.f;
    }
    c0 = wmma4(a, b0, c0);
    c1 = wmma4(a, b1, c1);
  }
  float bo0 = ws[bOff + row16];
  float bo1 = (row16 < 8) ? ws[bOff + 16 + row16] : 0.f;
  int r0 = half * 8;
#pragma unroll
  for (int v = 0; v < 8; ++v) {
    float t0 = c0[v] + bo0;
    float t1 = c1[v] + bo1;
    if (relu) { t0 = fmaxf(t0, 0.f); t1 = fmaxf(t1, 0.f); }
    mxa[v] = fmaxf(mxa[v], t0);
    mxb[v] = fmaxf(mxb[v], t1);
    if (store) {
      Y[(r0 + v) * LR + storeOff + row16] = t0;
      if (row16 < 8) Y[(r0 + v) * LR + storeOff + 16 + row16] = t1;
    }
  }
}

// ---------------------------------------------------------------------------
// fused dense (EdgeConv) block, WMMA version. Block = 16 points, 8 waves.
// Wave w owns neighbor slots {w, w+8, (+16 for wave 0)}; per-wave LDS tile
// Y[16][176] holds [h3|h2|h1|ctr|diff] rows; layers chain through Y.
// Cross-wave max via LDS partials + reduction pass.
// Output channels: [h3(24), h2(24), h1(24), ctr(C), diffmax(C)].
// ---------------------------------------------------------------------------
__global__ __launch_bounds__(256) void k_dense(const float* __restrict__ f, int C,
                                               const int* __restrict__ idx,
                                               const float* __restrict__ w1, const float* __restrict__ b1,
                                               const float* __restrict__ w2, const float* __restrict__ b2,
                                               const float* __restrict__ w3, const float* __restrict__ b3,
                                               float* __restrict__ dst,
                                               long long rowStride, long long chStride) {
  __shared__ float ws[8712];
  __shared__ float yw[8][16 * 176];
  const int LR  = 176;
  const int C2  = 2 * C;
  const int o2  = 24 * C2;
  const int o3  = o2 + 24 * (C2 + 24);
  const int ob1 = o3 + 24 * (C2 + 48);
  const int ob2 = ob1 + 24;
  const int ob3 = ob2 + 24;

#ifdef USE_ASYNC_LDS
  for (int t = threadIdx.x; t < 24 * C2; t += 256)
    __builtin_amdgcn_global_load_async_to_lds_b32(
        (as1_int*)(w1 + t), (as3_int*)(ws + t), 0, 0);
  for (int t = threadIdx.x; t < 24 * (C2 + 24); t += 256)
    __builtin_amdgcn_global_load_async_to_lds_b32(
        (as1_int*)(w2 + t), (as3_int*)(ws + o2 + t), 0, 0);
  for (int t = threadIdx.x; t < 24 * (C2 + 48); t += 256)
    __builtin_amdgcn_global_load_async_to_lds_b32(
        (as1_int*)(w3 + t), (as3_int*)(ws + o3 + t), 0, 0);
  __builtin_amdgcn_s_wait_asynccnt(0);
#else
  for (int t = threadIdx.x; t < 24 * C2; t += 256)        ws[t]      = w1[t];
  for (int t = threadIdx.x; t < 24 * (C2 + 24); t += 256) ws[o2 + t] = w2[t];
  for (int t = threadIdx.x; t < 24 * (C2 + 48); t += 256) ws[o3 + t] = w3[t];
#endif
  if (threadIdx.x < 24) {
    ws[ob1 + threadIdx.x] = b1[threadIdx.x];
    ws[ob2 + threadIdx.x] = b2[threadIdx.x];
    ws[ob3 + threadIdx.x] = b3[threadIdx.x];
  }
  __syncthreads();

  const int wave  = threadIdx.x >> 5;
  const int lane  = threadIdx.x & 31;
  const int row16 = lane & 15;
  const int half  = lane >> 4;
  const int koff  = half << 1;
  const int pbase = blockIdx.x * 16;
  float* Y = yw[wave];

  // stage ctr channels (each wave keeps its own copy)
  const int ch0 = half * (C >> 1), ch1 = (half + 1) * (C >> 1);
  for (int c = ch0; c < ch1; ++c)
    Y[row16 * LR + 72 + c] = f[(size_t)(pbase + row16) * C + c];

  v8f mh1a = splat8(-3.4e38f), mh1b = splat8(-3.4e38f);
  v8f mh2a = splat8(-3.4e38f), mh2b = splat8(-3.4e38f);
  v8f mh3a = splat8(-3.4e38f), mh3b = splat8(-3.4e38f);

  for (int kk = wave; kk < KK; kk += 8) {
    int j = idx[(pbase + row16) * KK + kk];
    for (int c = ch0; c < ch1; ++c)
      Y[row16 * LR + 72 + C + c] = f[(size_t)j * C + c] - Y[row16 * LR + 72 + c];

    edge_layer(Y, ws, 0,  ob1, 72, C2,      LR, row16, half, koff, true,  true, 48, mh1a, mh1b);
    edge_layer(Y, ws, o2, ob2, 48, C2 + 24, LR, row16, half, koff, true,  true, 24, mh2a, mh2b);
    edge_layer(Y, ws, o3, ob3, 24, C2 + 48, LR, row16, half, koff, false, false, 0, mh3a, mh3b);
  }

  // write per-wave partial maxima for the h-channels into own Y tile (ch 0..71)
#pragma unroll
  for (int v = 0; v < 8; ++v) {
    int r = half * 8 + v;
    Y[r * LR + 0  + row16] = mh3a[v];
    Y[r * LR + 24 + row16] = mh2a[v];
    Y[r * LR + 48 + row16] = mh1a[v];
    if (row16 < 8) {
      Y[r * LR + 16 + row16] = mh3b[v];
      Y[r * LR + 40 + row16] = mh2b[v];
      Y[r * LR + 64 + row16] = mh1b[v];
    }
  }
  __syncthreads();

  // reduce across waves: h channels
  for (int i = threadIdx.x; i < 16 * 72; i += 256) {
    int r = i / 72, ch = i % 72;
    float m = yw[0][r * LR + ch];
#pragma unroll
    for (int w = 1; w < 8; ++w) m = fmaxf(m, yw[w][r * LR + ch]);
    dst[(size_t)(pbase + r) * rowStride + (size_t)ch * chStride] = m;
  }
  // ctr + diff-max channels (elementwise, no GEMM content)
  for (int i = threadIdx.x; i < 16 * C; i += 256) {
    int r = i / C, c = i % C;
    int p = pbase + r;
    float fc = f[(size_t)p * C + c];
    dst[(size_t)p * rowStride + (size_t)(72 + c) * chStride] = fc;
    float md = -3.4e38f;
    for (int k = 0; k < KK; ++k) {
      int j = idx[p * KK + k];
      md = fmaxf(md, f[(size_t)j * C + c] - fc);
    }
    dst[(size_t)p * rowStride + (size_t)(72 + C + c) * chStride] = md;
  }
}

// dst[r][c] (stride dstS) = src[r][c] (stride srcC)
__global__ __launch_bounds__(256) void k_concat(const float* __restrict__ src,
                                                float* __restrict__ dst,
                                                int srcC, int dstS) {
  int t = blockIdx.x * 256 + threadIdx.x;
  if (t >= NN * srcC) return;
  int r = t / srcC, c = t % srcC;
  dst[(size_t)r * dstS + c] = src[(size_t)r * srcC + c];
}

// out[c*N + n] = src[n*C + c]
__global__ __launch_bounds__(256) void k_tcopy(const float* __restrict__ src,
                                               float* __restrict__ out, int C) {
  int t = blockIdx.x * 256 + threadIdx.x;
  if (t >= NN * C) return;
  int n = t / C, c = t % C;
  out[(size_t)c * NN + n] = src[(size_t)n * C + c];
}

// ---------------------------------------------------------------------------
extern "C" void kernel_launch(void* const* d_in, const int* in_sizes, int n_in,
                              void* d_out, int out_size, void* d_ws, size_t ws_size,
                              hipStream_t stream) {
  (void)in_sizes; (void)n_in; (void)out_size; (void)ws_size;
  const float* x   = (const float*)d_in[0];
  const float* c1w = (const float*)d_in[1];
  const float* c1b = (const float*)d_in[2];
  const float *dW[4][3], *dBi[4][3];
  int base = 3;
  for (int l = 0; l < 4; ++l)
    for (int m = 0; m < 3; ++m) {
      dW[l][m]  = (const float*)d_in[base++];
      dBi[l][m] = (const float*)d_in[base++];
    }
  const float* c2w = (const float*)d_in[27];
  const float* c2b = (const float*)d_in[28];
  const float* c3w = (const float*)d_in[29];
  const float* c3b = (const float*)d_in[30];
  const float* c4w = (const float*)d_in[31];
  const float* c4b = (const float*)d_in[32];
  float* out = (float*)d_out;

  float* W = (float*)d_ws;
  size_t off = 0;
  auto alloc = [&](size_t n) { float* p = W + off; off += n; return p; };
  float* l0 = alloc((size_t)BB * NN * 24);
  float* l1 = alloc((size_t)BB * NN * 144);
  float* l2 = alloc((size_t)BB * NN * 312);
  float* l3 = alloc((size_t)BB * NN * 480);
  float* fbuf = alloc((size_t)NN * 48);
  float* G  = alloc((size_t)NN * NN);
  float* sq = alloc((size_t)NN);
  int*  idx = (int*)alloc((size_t)NN * KK);

  k_c1<<<(BB * NN + 255) / 256, 256, 0, stream>>>(x, c1w, c1b, l0);

  for (int b = 0; b < BB; ++b) {
    float* l0b = l0 + (size_t)b * NN * 24;
    float* l1b = l1 + (size_t)b * NN * 144;
    float* l2b = l2 + (size_t)b * NN * 312;
    float* l3b = l3 + (size_t)b * NN * 480;
    float* outb = out + (size_t)b * 648 * NN;

    // ---- layer 1: f = l0 (C=24) -> l1 = [dense(120) | l0(24)]
    k_sqnorm<<<NN / 256, 256, 0, stream>>>(l0b, sq, 24);
    k_gram<<<dim3(8, 256), 256, 0, stream>>>(l0b, G, 24);
    k_topk<<<NN / 256, 256, 0, stream>>>(G, sq, idx);
    k_dense<<<NN / 16, 256, 0, stream>>>(l0b, 24, idx,
        dW[0][0], dBi[0][0], dW[0][1], dBi[0][1], dW[0][2], dBi[0][2],
        l1b, 144, 1);
    k_concat<<<(NN * 24 + 255) / 256, 256, 0, stream>>>(l0b, l1b + 120, 24, 144);

    // ---- layer 2: f = relu(c2(l1)) (C=48) -> l2 = [dense(168)|l1(144)]
    k_lin48<<<dim3(3, 32), 256, 0, stream>>>(l1b, c2w, c2b, fbuf, 144);
    k_sqnorm<<<NN / 256, 256, 0, stream>>>(fbuf, sq, 48);
    k_gram<<<dim3(8, 256), 256, 0, stream>>>(fbuf, G, 48);
    k_topk<<<NN / 256, 256, 0, stream>>>(G, sq, idx);
    k_dense<<<NN / 16, 256, 0, stream>>>(fbuf, 48, idx,
        dW[1][0], dBi[1][0], dW[1][1], dBi[1][1], dW[1][2], dBi[1][2],
        l2b, 312, 1);
    k_concat<<<(NN * 144 + 255) / 256, 256, 0, stream>>>(l1b, l2b + 168, 144, 312);

    // ---- layer 3
    k_lin48<<<dim3(3, 32), 256, 0, stream>>>(l2b, c3w, c3b, fbuf, 312);
    k_sqnorm<<<NN / 256, 256, 0, stream>>>(fbuf, sq, 48);
    k_gram<<<dim3(8, 256), 256, 0, stream>>>(fbuf, G, 48);
    k_topk<<<NN / 256, 256, 0, stream>>>(G, sq, idx);
    k_dense<<<NN / 16, 256, 0, stream>>>(fbuf, 48, idx,
        dW[2][0], dBi[2][0], dW[2][1], dBi[2][1], dW[2][2], dBi[2][2],
        l3b, 480, 1);
    k_concat<<<(NN * 312 + 255) / 256, 256, 0, stream>>>(l2b, l3b + 168, 312, 480);

    // ---- layer 4: dense -> out[b][0:168][n] (transposed), l3 -> out[b][168:648][n]
    k_lin48<<<dim3(3, 32), 256, 0, stream>>>(l3b, c4w, c4b, fbuf, 480);
    k_sqnorm<<<NN / 256, 256, 0, stream>>>(fbuf, sq, 48);
    k_gram<<<dim3(8, 256), 256, 0, stream>>>(fbuf, G, 48);
    k_topk<<<NN / 256, 256, 0, stream>>>(G, sq, idx);
    k_dense<<<NN / 16, 256, 0, stream>>>(fbuf, 48, idx,
        dW[3][0], dBi[3][0], dW[3][1], dBi[3][1], dW[3][2], dBi[3][2],
        outb, 1, NN);
    k_tcopy<<<(NN * 480 + 255) / 256, 256, 0, stream>>>(l3b, outb + (size_t)168 * NN, 480);
  }
}